// QuantumFeedForward_65481071406287
// MI455X (gfx1250) — compile-verified
//
#include <hip/hip_runtime.h>

typedef __attribute__((ext_vector_type(2))) float v2f;
typedef __attribute__((ext_vector_type(8))) float v8f;

#define EMBED_DIM 512
#define FFN_DIM   2048
#define QUBITS    8
#define WAVES_PER_BLOCK   8   // 256 threads, wave32
#define TOKTILES_PER_WAVE 8   // 16-token tiles processed per wave

// out[t,f] = sum_q cos(x[t,q]) * cos(params[q,0]) * W[f,q] + b[f]
// Computed as D = A(16x4 f32) x B(4x16 f32) + C via V_WMMA_F32_16X16X4_F32, K=8 -> 2 WMMAs.
__global__ __launch_bounds__(256, 1) void QuantumFeedForward_wmma_kernel(
    const float* __restrict__ x, const float* __restrict__ params,
    const float* __restrict__ W, const float* __restrict__ bias,
    float* __restrict__ out)
{
    const int lane = threadIdx.x & 31;
    const int wave = threadIdx.x >> 5;
    const int hi   = lane >> 4;    // 0: lanes 0-15, 1: lanes 16-31
    const int l15  = lane & 15;
    const int kb   = hi * 2;       // K pair base within a 4-wide K block

    // f-tile owned by this wave (16 output columns)
    const int ftile = blockIdx.y * WAVES_PER_BLOCK + wave;
    const int f     = ftile * 16 + l15;

    // Fold cos(params[q,0]) into the B matrix (loaded once per wave).
    const float cp0 = __cosf(params[(kb + 0) * 3]);
    const float cp1 = __cosf(params[(kb + 1) * 3]);
    const float cp4 = __cosf(params[(kb + 4) * 3]);
    const float cp5 = __cosf(params[(kb + 5) * 3]);

    // B layout (4x16, 32-bit): VGPR0 = K=kb row, VGPR1 = K=kb+1 row, N = l15.
    const float* wrow = W + (size_t)f * QUBITS;   // W is [FFN, QUBITS] row-major
    v2f B0, B1;
    B0.x = wrow[kb + 0] * cp0;  // WMMA0: K = 0..3
    B0.y = wrow[kb + 1] * cp1;
    B1.x = wrow[kb + 4] * cp4;  // WMMA1: K = 4..7
    B1.y = wrow[kb + 5] * cp5;

    // Bias baked into the accumulator init (C/D: VGPR j -> M = j + 8*hi, N = l15).
    const float bf = bias[f];
    v8f cinit;
#pragma unroll
    for (int j = 0; j < 8; ++j) cinit[j] = bf;

    const int ttile0 = blockIdx.x * TOKTILES_PER_WAVE;
#pragma unroll 1
    for (int i = 0; i < TOKTILES_PER_WAVE; ++i) {
        const int tbase = (ttile0 + i) * 16;

        // A layout (16x4, 32-bit): lane half selects K pair {kb, kb+1}; M = l15.
        const float* xp = x + (size_t)(tbase + l15) * EMBED_DIM;
        v2f A0, A1;
        A0.x = __cosf(xp[kb + 0]);
        A0.y = __cosf(xp[kb + 1]);
        A1.x = __cosf(xp[kb + 4]);
        A1.y = __cosf(xp[kb + 5]);

        v8f acc = cinit;
        // 8 args: (neg_a, A, neg_b, B, c_mod, C, reuse_a, reuse_b)
        acc = __builtin_amdgcn_wmma_f32_16x16x4_f32(false, A0, false, B0,
                                                    (short)0, acc, false, false);
        acc = __builtin_amdgcn_wmma_f32_16x16x4_f32(false, A1, false, B1,
                                                    (short)0, acc, false, false);

        // Store 16x16 tile: lane writes rows tbase + 8*hi + j, column f.
        // 512 MB streaming output >> 192 MB L2 -> non-temporal stores.
        float* op = out + (size_t)(tbase + hi * 8) * FFN_DIM + f;
#pragma unroll
        for (int j = 0; j < 8; ++j) {
            __builtin_nontemporal_store(acc[j], op);
            op += FFN_DIM;
        }
    }
}

extern "C" void kernel_launch(void* const* d_in, const int* in_sizes, int n_in,
                              void* d_out, int out_size, void* d_ws, size_t ws_size,
                              hipStream_t stream) {
    const float* x      = (const float*)d_in[0];   // [16, 4096, 512] f32
    const float* params = (const float*)d_in[1];   // [8, 3] f32
    const float* W      = (const float*)d_in[2];   // [2048, 8] f32
    const float* b      = (const float*)d_in[3];   // [2048] f32
    float* out          = (float*)d_out;           // [16, 4096, 2048] f32

    const int ntok = in_sizes[0] / EMBED_DIM;      // 65536 tokens
    dim3 grid(ntok / (16 * TOKTILES_PER_WAVE),     // 512 token-tile groups
              FFN_DIM / (16 * WAVES_PER_BLOCK));   // 16 f-tile groups
    QuantumFeedForward_wmma_kernel<<<grid, dim3(256), 0, stream>>>(x, params, W, b, out);
}